// VanillaQGNN_90280212562549
// MI455X (gfx1250) — compile-verified
//
#include <hip/hip_runtime.h>
#include <hip/hip_bf16.h>

typedef __attribute__((ext_vector_type(2))) float v2f;
typedef __attribute__((ext_vector_type(8))) float v8f;

#define HALF_PI_F 1.57079632679489662f

// ---------------------------------------------------------------------------
// Fused GEMM: O[N x NC] = act( A[N x 128] @ W[128 x NC] + bias
//                              (+ qv[N x 4] @ Wq[4 x NC] + bq)  if QADD )
// NC = NT*16. One 256-thread block (8 waves) covers 128 rows; each wave owns a
// 16-row strip and NT 16x16 accumulator tiles. W staged in LDS (<=64KB).
// Uses V_WMMA_F32_16X16X4_F32 (fp32-exact on CDNA5).
// ---------------------------------------------------------------------------
template <int NT, bool RELU, bool QADD>
__global__ __launch_bounds__(256) void k_gemm_wmma(
    const float* __restrict__ A,     // [N,128] row-major
    const float* __restrict__ W,     // [128,NC] row-major
    const float* __restrict__ bias,  // [NC]
    const float* __restrict__ qv,    // [N,4]   (QADD only)
    const float* __restrict__ Wq,    // [4,NC]  (QADD only)
    const float* __restrict__ bq,    // [NC]    (QADD only)
    float* __restrict__ O,           // [N,NC]
    int n)
{
    constexpr int NC = NT * 16;
    __shared__ float sW[128 * NC];
    for (int i = threadIdx.x; i < 128 * NC; i += 256) sW[i] = W[i];
    __syncthreads();

    const int lane = threadIdx.x & 31;
    const int wave = threadIdx.x >> 5;
    const int hi   = lane >> 4;   // 0: K pair {0,1}; 1: K pair {2,3}
    const int ln   = lane & 15;   // M (A frag) / N (B,C,D frags)
    const int base = blockIdx.x * 128 + wave * 16;
    if (base >= n) return;        // wave-uniform: EXEC stays all-ones for WMMA

    // clamp A row for tail (stores are predicated; loads must stay uniform)
    int rA = base + ln;
    if (rA >= n) rA = n - 1;
    const float* __restrict__ arow = A + (size_t)rA * 128;

    v8f acc[NT];
#pragma unroll
    for (int t = 0; t < NT; ++t) {
        float bv = bias[t * 16 + ln];
        if (QADD) bv += bq[t * 16 + ln];
#pragma unroll
        for (int j = 0; j < 8; ++j) acc[t][j] = bv;
    }

#pragma unroll 4
    for (int kk = 0; kk < 32; ++kk) {
        const int k = kk * 4 + hi * 2;
        v2f a;
        a.x = arow[k];
        a.y = arow[k + 1];
#pragma unroll
        for (int t = 0; t < NT; ++t) {
            v2f b;
            b.x = sW[(k + 0) * NC + t * 16 + ln];
            b.y = sW[(k + 1) * NC + t * 16 + ln];
            acc[t] = __builtin_amdgcn_wmma_f32_16x16x4_f32(
                false, a, false, b, (short)0, acc[t], false, false);
        }
    }

    if (QADD) {
        const int k = hi * 2;
        v2f a;
        a.x = qv[(size_t)rA * 4 + k];
        a.y = qv[(size_t)rA * 4 + k + 1];
#pragma unroll
        for (int t = 0; t < NT; ++t) {
            v2f b;
            b.x = Wq[(k + 0) * NC + t * 16 + ln];
            b.y = Wq[(k + 1) * NC + t * 16 + ln];
            acc[t] = __builtin_amdgcn_wmma_f32_16x16x4_f32(
                false, a, false, b, (short)0, acc[t], false, false);
        }
    }

#pragma unroll
    for (int t = 0; t < NT; ++t) {
#pragma unroll
        for (int r = 0; r < 8; ++r) {
            const int row = base + r + hi * 8;
            if (row < n) {
                float v = acc[t][r];
                if (RELU) v = fmaxf(v, 0.0f);
                O[(size_t)row * NC + t * 16 + ln] = v;
            }
        }
    }
}

// ---------------------------------------------------------------------------
// Quantum features: qv[i,q] = cumprod_q( cos(th)cos(xn) - sin(th)sin(xn)cos(phi) )
// with xn = tanh(h[i,q]) * pi/2, q = 0..3.  qp layout: [4][3] (phi,theta,_).
// ---------------------------------------------------------------------------
__global__ void k_quantum(const float* __restrict__ h,
                          const float* __restrict__ qp,
                          float* __restrict__ qv, int n)
{
    int i = blockIdx.x * blockDim.x + threadIdx.x;
    if (i >= n) return;
    float c = 1.0f;
#pragma unroll
    for (int q = 0; q < 4; ++q) {
        float xn    = tanhf(h[(size_t)i * 128 + q]) * HALF_PI_F;
        float phi   = qp[q * 3 + 0];
        float theta = qp[q * 3 + 1];
        float z = cosf(theta) * cosf(xn) - sinf(theta) * sinf(xn) * cosf(phi);
        c *= z;
        qv[(size_t)i * 4 + q] = c;
    }
}

// agg = comb  (realizes the self-loop term of segment_sum), float4-wide
__global__ void k_copy4(const float4* __restrict__ s, float4* __restrict__ d, int n4)
{
    int i = blockIdx.x * blockDim.x + threadIdx.x;
    if (i < n4) d[i] = s[i];
}

// agg[dst[e]] += comb[src[e]]  over the 800k real edges; 1 thread = 4 floats
__global__ void k_scatter(const float* __restrict__ comb,
                          const int* __restrict__ ei,  // [2,E]
                          float* __restrict__ agg, int E)
{
    int tid = blockIdx.x * blockDim.x + threadIdx.x;
    int e = tid >> 5;
    if (e >= E) return;
    int j = (tid & 31) * 4;
    int s = ei[e];
    int d = ei[E + e];
    float4 v = *(const float4*)&comb[(size_t)s * 128 + j];
    float* o = &agg[(size_t)d * 128 + j];
    atomicAdd(o + 0, v.x);
    atomicAdd(o + 1, v.y);
    atomicAdd(o + 2, v.z);
    atomicAdd(o + 3, v.w);
}

// h += relu(agg), float4-wide
__global__ void k_update(float* __restrict__ h, const float* __restrict__ agg, int n4)
{
    int i = blockIdx.x * blockDim.x + threadIdx.x;
    if (i >= n4) return;
    float4 a  = ((const float4*)agg)[i];
    float4 hv = ((float4*)h)[i];
    hv.x += fmaxf(a.x, 0.0f);
    hv.y += fmaxf(a.y, 0.0f);
    hv.z += fmaxf(a.z, 0.0f);
    hv.w += fmaxf(a.w, 0.0f);
    ((float4*)h)[i] = hv;
}

extern "C" void kernel_launch(void* const* d_in, const int* in_sizes, int n_in,
                              void* d_out, int out_size, void* d_ws, size_t ws_size,
                              hipStream_t stream)
{
    const float* x       = (const float*)d_in[0];   // [N,128]
    const int*   ei      = (const int*)  d_in[1];   // [2,E]
    const float* W_in    = (const float*)d_in[2];   // [128,128]
    const float* b_in    = (const float*)d_in[3];   // [128]
    const float* W_lin   = (const float*)d_in[4];   // [2,128,128]
    const float* b_lin   = (const float*)d_in[5];   // [2,128]
    const float* qparams = (const float*)d_in[6];   // [2,4,3]
    const float* W_qp    = (const float*)d_in[7];   // [2,4,128]
    const float* b_qp    = (const float*)d_in[8];   // [2,128]
    const float* W_out   = (const float*)d_in[9];   // [128,64]
    const float* b_out   = (const float*)d_in[10];  // [64]

    const int N = in_sizes[0] / 128;
    const int E = in_sizes[1] / 2;

    float* h    = (float*)d_ws;                     // [N,128]
    float* comb = h    + (size_t)N * 128;           // [N,128]
    float* agg  = comb + (size_t)N * 128;           // [N,128]
    float* qv   = agg  + (size_t)N * 128;           // [N,4]

    const int gblk  = (N + 127) / 128;              // gemm blocks (128 rows each)
    const int eblk  = (N + 255) / 256;              // elementwise over nodes
    const int v4n   = N * 32;                       // float4 count of [N,128]
    const int v4blk = (v4n + 255) / 256;
    const int sblk  = (E * 32 + 255) / 256;         // scatter threads

    // h = relu(x @ W_in + b_in)
    k_gemm_wmma<8, true, false><<<gblk, 256, 0, stream>>>(
        x, W_in, b_in, nullptr, nullptr, nullptr, h, N);

    for (int l = 0; l < 2; ++l) {
        k_quantum<<<eblk, 256, 0, stream>>>(h, qparams + l * 12, qv, N);

        // comb = h @ W_lin[l] + b_lin[l] + qv @ W_qp[l] + b_qp[l]
        k_gemm_wmma<8, false, true><<<gblk, 256, 0, stream>>>(
            h, W_lin + (size_t)l * 128 * 128, b_lin + l * 128,
            qv, W_qp + (size_t)l * 4 * 128, b_qp + l * 128, comb, N);

        // agg = comb (self-loops), then scatter real edges
        k_copy4<<<v4blk, 256, 0, stream>>>((const float4*)comb, (float4*)agg, v4n);
        k_scatter<<<sblk, 256, 0, stream>>>(comb, ei, agg, E);

        // h = h + relu(agg)
        k_update<<<v4blk, 256, 0, stream>>>(h, agg, v4n);
    }

    // out = h @ W_out + b_out
    k_gemm_wmma<4, false, false><<<gblk, 256, 0, stream>>>(
        h, W_out, b_out, nullptr, nullptr, nullptr, (float*)d_out, N);
}